// GCN_53919019434049
// MI455X (gfx1250) — compile-verified
//
#include <hip/hip_runtime.h>

typedef __attribute__((ext_vector_type(2))) float v2f;
typedef __attribute__((ext_vector_type(8))) float v8f;

static __device__ __forceinline__ void atomic_add_f32(float* p, float v) {
  __hip_atomic_fetch_add(p, v, __ATOMIC_RELAXED, __HIP_MEMORY_SCOPE_AGENT);
}

// ---------------- degree / norm ----------------
__global__ void k_init_deg(float* deg, int n) {
  int i = blockIdx.x * blockDim.x + threadIdx.x;
  if (i < n) deg[i] = 1.0f;  // self-loop contributes 1
}

__global__ void k_count_deg(const int* __restrict__ dst, float* deg, int E) {
  int i = blockIdx.x * blockDim.x + threadIdx.x;
  if (i < E) atomic_add_f32(&deg[dst[i]], 1.0f);
}

__global__ void k_rsqrt_deg(float* d, int n) {
  int i = blockIdx.x * blockDim.x + threadIdx.x;
  if (i < n) d[i] = rsqrtf(d[i]);  // deg >= 1 always
}

// ---------------- WMMA GEMM: out[nrows x NOUT] = (relu?)A[nrows x 128] @ W[128 x NOUT] ----------------
// W staged in LDS with k row-pairs interleaved so each B fragment is one ds_load_b64:
//   Wl[(k>>1)*(2*NOUT) + 2*j + (k&1)] = W[k][j]
template <int NOUT, bool RELU>
__global__ __launch_bounds__(256) void k_gemm_wmma(const float* __restrict__ A,
                                                   const float* __restrict__ W,
                                                   float* __restrict__ out, int nrows) {
  __shared__ float Wl[128 * NOUT];
  const int tid = threadIdx.x;
  for (int i = tid; i < 128 * NOUT; i += 256) {
    int k = i / NOUT;
    int j = i - k * NOUT;
    Wl[(k >> 1) * (2 * NOUT) + 2 * j + (k & 1)] = W[i];
  }
  __syncthreads();

  const int wave = tid >> 5;
  const int lane = tid & 31;
  const int tile = blockIdx.x * 8 + wave;  // 16-row strip per wave
  const int ntiles = (nrows + 15) >> 4;
  if (tile >= ntiles) return;  // wave-uniform: EXEC stays all-ones for WMMA

  const int m = lane & 15;   // A row / B col within tile
  const int kh = lane >> 4;  // half-wave selects k-pair {0,1} vs {2,3}
  int row = tile * 16 + m;
  if (row >= nrows) row = nrows - 1;  // clamped duplicate load (stores guarded)

  v8f acc[NOUT / 16];
#pragma unroll
  for (int t = 0; t < NOUT / 16; ++t) acc[t] = v8f{0.f, 0.f, 0.f, 0.f, 0.f, 0.f, 0.f, 0.f};

  const float* arow = A + (size_t)row * 128 + 2 * kh;
#pragma unroll 4
  for (int k0 = 0; k0 < 128; k0 += 4) {
    v2f a = *(const v2f*)(arow + k0);  // {A[m][k0+2kh], A[m][k0+2kh+1]}
    if (RELU) {
      a.x = fmaxf(a.x, 0.f);
      a.y = fmaxf(a.y, 0.f);
    }
    const float* bb = &Wl[((k0 >> 1) + kh) * (2 * NOUT)];
#pragma unroll
    for (int t = 0; t < NOUT / 16; ++t) {
      v2f b = *(const v2f*)(bb + 2 * (t * 16 + m));  // {W[k][n], W[k+1][n]}
      acc[t] = __builtin_amdgcn_wmma_f32_16x16x4_f32(false, a, false, b, (short)0, acc[t],
                                                     false, false);
    }
  }

  // C layout: lanes 0-15 -> rows tile*16 + v, lanes 16-31 -> rows tile*16 + 8 + v; col = t*16 + m
  const int rbase = tile * 16 + kh * 8;
#pragma unroll
  for (int t = 0; t < NOUT / 16; ++t) {
#pragma unroll
    for (int v = 0; v < 8; ++v) {
      int r = rbase + v;
      if (r < nrows) out[(size_t)r * NOUT + t * 16 + m] = acc[t][v];
    }
  }
}

// ---------------- aggregation init: agg = h * dinv^2 (self loop) + bias ----------------
template <int NOUT>
__global__ void k_init_agg(const float* __restrict__ h, const float* __restrict__ dinv,
                           const float* __restrict__ bias, float* __restrict__ agg, int n) {
  int i = blockIdx.x * blockDim.x + threadIdx.x;
  if (i >= n * NOUT) return;
  int r = i / NOUT;
  int c = i - r * NOUT;
  float di = dinv[r];
  agg[i] = h[i] * di * di + bias[c];
}

// ---------------- edge scatter, width 128: one wave per edge ----------------
__global__ __launch_bounds__(256) void k_scatter128(const float* __restrict__ h,
                                                    const int* __restrict__ src,
                                                    const int* __restrict__ dst,
                                                    const float* __restrict__ dinv,
                                                    float* __restrict__ agg, int E) {
  int gw = (blockIdx.x * 256 + threadIdx.x) >> 5;
  int lane = threadIdx.x & 31;
  if (gw >= E) return;
  int s = src[gw], d = dst[gw];
  float nrm = dinv[s] * dinv[d];
  const float4 hv = *(const float4*)(h + (size_t)s * 128 + lane * 4);
  float* o = agg + (size_t)d * 128 + lane * 4;
  atomic_add_f32(o + 0, hv.x * nrm);
  atomic_add_f32(o + 1, hv.y * nrm);
  atomic_add_f32(o + 2, hv.z * nrm);
  atomic_add_f32(o + 3, hv.w * nrm);
}

// ---------------- edge scatter, width 64: one half-wave per edge ----------------
__global__ __launch_bounds__(256) void k_scatter64(const float* __restrict__ h,
                                                   const int* __restrict__ src,
                                                   const int* __restrict__ dst,
                                                   const float* __restrict__ dinv,
                                                   float* __restrict__ out, int E) {
  int t = blockIdx.x * 256 + threadIdx.x;
  int e = t >> 4;
  int lane = threadIdx.x & 15;
  if (e >= E) return;
  int s = src[e], d = dst[e];
  float nrm = dinv[s] * dinv[d];
  const float4 hv = *(const float4*)(h + (size_t)s * 64 + lane * 4);
  float* o = out + (size_t)d * 64 + lane * 4;
  atomic_add_f32(o + 0, hv.x * nrm);
  atomic_add_f32(o + 1, hv.y * nrm);
  atomic_add_f32(o + 2, hv.z * nrm);
  atomic_add_f32(o + 3, hv.w * nrm);
}

extern "C" void kernel_launch(void* const* d_in, const int* in_sizes, int n_in, void* d_out,
                              int out_size, void* d_ws, size_t ws_size, hipStream_t stream) {
  const float* x = (const float*)d_in[0];
  const int* ei = (const int*)d_in[1];
  const float* W1 = (const float*)d_in[2];
  const float* b1 = (const float*)d_in[3];
  const float* W2 = (const float*)d_in[4];
  const float* b2 = (const float*)d_in[5];
  float* out = (float*)d_out;

  const int N = in_sizes[0] / 128;  // 50000
  const int E = in_sizes[1] / 2;    // 800000
  const int* src = ei;
  const int* dstp = ei + E;

  // workspace layout: dinv | h1 (reused as h2) | agg1   (~51.4 MB)
  char* ws = (char*)d_ws;
  float* dinv = (float*)ws;
  size_t off = (((size_t)N * 4) + 255) & ~(size_t)255;
  float* h1 = (float*)(ws + off);
  float* agg1 = (float*)(ws + off + (size_t)N * 128 * 4);
  float* h2 = h1;  // h1 dead after agg1 is built

  dim3 B(256);
  k_init_deg<<<(N + 255) / 256, B, 0, stream>>>(dinv, N);
  k_count_deg<<<(E + 255) / 256, B, 0, stream>>>(dstp, dinv, E);
  k_rsqrt_deg<<<(N + 255) / 256, B, 0, stream>>>(dinv, N);

  const int ntiles = (N + 15) / 16;
  const int gblocks = (ntiles + 7) / 8;

  // Layer 1: h1 = x @ W1 ; agg1 = scatter(h1) + b1 ; relu fused into layer-2 GEMM load
  k_gemm_wmma<128, false><<<gblocks, B, 0, stream>>>(x, W1, h1, N);
  k_init_agg<128><<<(N * 128 + 255) / 256, B, 0, stream>>>(h1, dinv, b1, agg1, N);
  k_scatter128<<<(E + 7) / 8, B, 0, stream>>>(h1, src, dstp, dinv, agg1, E);

  // Layer 2: h2 = relu(agg1) @ W2 ; out = scatter(h2) + b2
  k_gemm_wmma<64, true><<<gblocks, B, 0, stream>>>(agg1, W2, h2, N);
  k_init_agg<64><<<(N * 64 + 255) / 256, B, 0, stream>>>(h2, dinv, b2, out, N);
  k_scatter64<<<(E + 15) / 16, B, 0, stream>>>(h2, src, dstp, dinv, out, E);
}